// TransformerBlock_8796093022943
// MI455X (gfx1250) — compile-verified
//
#include <hip/hip_runtime.h>

typedef float v2f __attribute__((ext_vector_type(2)));
typedef float v8f __attribute__((ext_vector_type(8)));
typedef unsigned int u32x4 __attribute__((ext_vector_type(4)));
typedef int i32x4 __attribute__((ext_vector_type(4)));
typedef int i32x8 __attribute__((ext_vector_type(8)));

#define S_LEN   4096
#define D_MODEL 768
#define F_FFN   2048
#define NHEAD   12
#define HDIM    64

static __device__ __forceinline__ v8f wmma4(v2f a, v2f b, v8f c) {
  // D = A(16x4,f32) * B(4x16,f32) + C(16x16,f32), full fp32 precision
  return __builtin_amdgcn_wmma_f32_16x16x4_f32(false, a, false, b, (short)0, c,
                                               false, false);
}

// ---------------------------------------------------------------------------
// Tensor Data Mover: 2-D tile load global -> LDS with hardware row padding.
// D# layout per CDNA5 ISA ch.8 (group0 128b, group1 256b; groups 2/3 zero).
//   tile_d0 = row length (elements, x-fastest), tile_d1 = #rows
//   stride0 = tensor row pitch (elements)
//   pad codes: interval 0=2,1=4,2=8,3=16,4=32,5=64 DWORDs; amount = code+1 DW
// ---------------------------------------------------------------------------
static __device__ __forceinline__ void tdm_load_2d(
    unsigned lds_addr, const void* gaddr, unsigned tensor_d0, unsigned tensor_d1,
    unsigned tile_d0, unsigned tile_d1, unsigned stride0,
    unsigned pad_interval_code, unsigned pad_amount_code) {
  unsigned long long ga = (unsigned long long)(size_t)gaddr;
  u32x4 g0;
  g0.x = 1u;                                            // count=1, user mode
  g0.y = lds_addr;                                      // LDS byte address
  g0.z = (unsigned)(ga & 0xFFFFFFFFull);                // global_addr[31:0]
  g0.w = (unsigned)((ga >> 32) & 0x1FFFFFFull) | (2u << 30);  // [56:32], type=2
  i32x8 g1;
  g1.s0 = (int)((2u << 16) |                            // data_size = 4 bytes
                (1u << 20) |                            // pad_enable
                (pad_interval_code << 22) | (pad_amount_code << 25));
  g1.s1 = (int)((tensor_d0 & 0xFFFFu) << 16);           // atomic_bar=0, dim0 lo
  g1.s2 = (int)(((tensor_d0 >> 16) & 0xFFFFu) | ((tensor_d1 & 0xFFFFu) << 16));
  g1.s3 = (int)(((tensor_d1 >> 16) & 0xFFFFu) | ((tile_d0 & 0xFFFFu) << 16));
  g1.s4 = (int)(tile_d1 & 0xFFFFu);                     // tile_dim1, tile_dim2=0
  g1.s5 = (int)stride0;                                 // dim0 stride lo32
  g1.s6 = 0;                                            // stride hi16, dim1_stride=0
  g1.s7 = 0;
  i32x4 z4 = {0, 0, 0, 0};
  i32x8 z8 = {0, 0, 0, 0, 0, 0, 0, 0};
  __builtin_amdgcn_tensor_load_to_lds(g0, g1, z4, z4, z8, 0);
}

static __device__ __forceinline__ unsigned lds_off(const void* p) {
  // generic LDS address: [63:32]=shared aperture, [31:0]=LDS byte address
  return (unsigned)(size_t)p;
}

// ---------------------------------------------------------------------------
// RMSNorm: one block per row of 768
// ---------------------------------------------------------------------------
__global__ __launch_bounds__(256) void rmsnorm_kernel(
    const float* __restrict__ x, const float* __restrict__ w,
    float* __restrict__ o) {
  const int row = blockIdx.x;
  const int tid = threadIdx.x;
  const float* xr = x + (size_t)row * D_MODEL;
  float s = 0.f;
  for (int i = tid; i < D_MODEL; i += 256) { float v = xr[i]; s += v * v; }
  __shared__ float red[256];
  red[tid] = s;
  __syncthreads();
  for (int st = 128; st > 0; st >>= 1) {
    if (tid < st) red[tid] += red[tid + st];
    __syncthreads();
  }
  const float scale = rsqrtf(red[0] * (1.0f / D_MODEL) + 1e-6f);
  float* orow = o + (size_t)row * D_MODEL;
  for (int i = tid; i < D_MODEL; i += 256) orow[i] = xr[i] * scale * w[i];
}

// ---------------------------------------------------------------------------
// Tiled fp32 WMMA GEMM with TDM tile fills: C[M,N] = A[M,K] @ B[K,N] (+ Res)
// Block tile 128x64x32; 8 waves 4(M)x2(N), each 32x32 = 2x2 WMMA accumulators.
// Wave w DMAs A rows [16w,16w+16) and B rows [4w,4w+4) of the tile.
// ---------------------------------------------------------------------------
__global__ __launch_bounds__(256) void gemm_wmma_f32(
    const float* __restrict__ A, const float* __restrict__ B,
    const float* __restrict__ Res, float* __restrict__ C,
    int M, int N, int K) {
  __shared__ float As[128][34];   // [m][k], pitch 34 (TDM pads 32+2 DW rows)
  __shared__ float Bs[32][68];    // [k][n], pitch 68 (TDM pads 64+4 DW rows)

  const int tid  = threadIdx.x;
  const int lane = tid & 31;
  const int wave = tid >> 5;
  const int hx   = lane >> 4;     // half-wave (K-pair select)
  const int lx   = lane & 15;     // M (for A) / N (for B,C)
  const int wm   = (wave >> 1) << 5;  // 0,32,64,96
  const int wn   = (wave & 1) << 5;   // 0,32
  const int m0   = blockIdx.y * 128;
  const int n0   = blockIdx.x * 64;

  const unsigned ldsA = lds_off(&As[0][0]) + (unsigned)(wave * 16 * 34 * 4);
  const unsigned ldsB = lds_off(&Bs[0][0]) + (unsigned)(wave * 4 * 68 * 4);
  const float* gA = A + (size_t)(m0 + wave * 16) * K;   // + k0 per iter
  const float* gB = B + (size_t)(wave * 4) * N + n0;    // + k0*N per iter

  v8f acc[2][2] = {};

  for (int k0 = 0; k0 < K; k0 += 32) {
    if (k0) __syncthreads();   // previous tile fully consumed
    // async DMA this wave's tile slices into LDS
    tdm_load_2d(ldsA, gA + k0, (unsigned)K, 1u << 20,
                32, 16, (unsigned)K, /*interval 32DW*/ 4, /*pad 2DW*/ 1);
    tdm_load_2d(ldsB, gB + (size_t)k0 * N, (unsigned)N, 1u << 20,
                64, 4, (unsigned)N, /*interval 64DW*/ 5, /*pad 4DW*/ 3);
    __builtin_amdgcn_s_wait_tensorcnt(0);
    __syncthreads();           // all 8 slices visible

#pragma unroll
    for (int kk = 0; kk < 32; kk += 4) {
      v2f a0 = *(const v2f*)&As[wm + lx][kk + 2 * hx];
      v2f a1 = *(const v2f*)&As[wm + 16 + lx][kk + 2 * hx];
      v2f b0, b1;
      b0.x = Bs[kk + 2 * hx][wn + lx];
      b0.y = Bs[kk + 2 * hx + 1][wn + lx];
      b1.x = Bs[kk + 2 * hx][wn + 16 + lx];
      b1.y = Bs[kk + 2 * hx + 1][wn + 16 + lx];
      acc[0][0] = wmma4(a0, b0, acc[0][0]);
      acc[0][1] = wmma4(a0, b1, acc[0][1]);
      acc[1][0] = wmma4(a1, b0, acc[1][0]);
      acc[1][1] = wmma4(a1, b1, acc[1][1]);
    }
  }

#pragma unroll
  for (int mi = 0; mi < 2; ++mi)
#pragma unroll
    for (int ni = 0; ni < 2; ++ni)
#pragma unroll
      for (int r = 0; r < 8; ++r) {
        const int row = m0 + wm + mi * 16 + r + 8 * hx;
        const int col = n0 + wn + ni * 16 + lx;
        float v = acc[mi][ni][r];
        if (Res) v += Res[(size_t)row * N + col];
        C[(size_t)row * N + col] = v;
      }
}

// ---------------------------------------------------------------------------
// RoPE in place on q and k: pair (j, j+32) per head dim, pos = seq index
// ---------------------------------------------------------------------------
__global__ __launch_bounds__(256) void rope_kernel(float* __restrict__ q,
                                                   float* __restrict__ k) {
  const int idx = blockIdx.x * blockDim.x + threadIdx.x;  // S*12*32
  const int j = idx & 31;
  const int h = (idx >> 5) % NHEAD;
  const int s = idx / (NHEAD * 32);
  const float inv = powf(10000.0f, -(float)j * (1.0f / 32.0f));
  const float f = (float)s * inv;
  const float c = cosf(f), sn = sinf(f);
  const size_t base = (size_t)s * D_MODEL + h * HDIM + j;
  float q0 = q[base], q1 = q[base + 32];
  q[base]      = q0 * c - q1 * sn;
  q[base + 32] = q1 * c + q0 * sn;
  float k0 = k[base], k1 = k[base + 32];
  k[base]      = k0 * c - k1 * sn;
  k[base + 32] = k1 * c + k0 * sn;
}

// ---------------------------------------------------------------------------
// Causal flash attention, fp32 WMMA. One wave owns 16 query rows.
// QK^T: 16 wmma per key tile; P@V: 16 wmma per key tile.
// ---------------------------------------------------------------------------
__global__ __launch_bounds__(256) void flash_attn_f32(
    const float* __restrict__ Q, const float* __restrict__ K,
    const float* __restrict__ V, float* __restrict__ O) {
  __shared__ float pbuf[8 * 16 * 18];   // per-wave P tile staging (C->A layout)
  const int tid  = threadIdx.x;
  const int lane = tid & 31;
  const int wave = tid >> 5;
  const int hx   = lane >> 4;
  const int lx   = lane & 15;
  const int head  = blockIdx.y;
  const int qtile = blockIdx.x * 8 + wave;
  const int qbase = qtile * 16;
  const int hoff  = head * HDIM;
  float* pb = pbuf + wave * 16 * 18;

  // Load Q fragments (16x64), pre-scaled by 1/sqrt(64)
  v2f qf[16];
#pragma unroll
  for (int kc = 0; kc < 16; ++kc) {
    v2f t = *(const v2f*)&Q[(size_t)(qbase + lx) * D_MODEL + hoff + kc * 4 + 2 * hx];
    qf[kc].x = t.x * 0.125f;
    qf[kc].y = t.y * 0.125f;
  }

  v8f oacc[4] = {};
  float mr[8], lr[8];
#pragma unroll
  for (int r = 0; r < 8; ++r) { mr[r] = -1e30f; lr[r] = 0.f; }

  for (int kt = 0; kt <= qtile; ++kt) {
    const int kbase = kt * 16;
    // S = Q @ K^T  (B fragment = K rows as columns)
    v8f s = {};
#pragma unroll
    for (int kc = 0; kc < 16; ++kc) {
      v2f b = *(const v2f*)&K[(size_t)(kbase + lx) * D_MODEL + hoff + kc * 4 + 2 * hx];
      s = wmma4(qf[kc], b, s);
    }
    // causal mask on diagonal tile
    if (kt == qtile) {
#pragma unroll
      for (int r = 0; r < 8; ++r) {
        const int row = qbase + r + 8 * hx;
        const int col = kbase + lx;
        if (col > row) s[r] = -3.0e38f;
      }
    }
    // online softmax (row = VGPR r within a 16-lane half)
    float p[8];
#pragma unroll
    for (int r = 0; r < 8; ++r) {
      float tm = s[r];
#pragma unroll
      for (int off = 8; off > 0; off >>= 1)
        tm = fmaxf(tm, __shfl_xor(tm, off, 32));
      const float nm = fmaxf(mr[r], tm);
      const float alpha = __expf(mr[r] - nm);
      mr[r] = nm;
      const float pv = __expf(s[r] - nm);
      p[r] = pv;
      float rs = pv;
#pragma unroll
      for (int off = 8; off > 0; off >>= 1)
        rs += __shfl_xor(rs, off, 32);
      lr[r] = lr[r] * alpha + rs;
#pragma unroll
      for (int nt = 0; nt < 4; ++nt) oacc[nt][r] *= alpha;
    }
    // stage P (C layout) -> LDS -> reread as A fragments (in-order DS, same wave)
#pragma unroll
    for (int r = 0; r < 8; ++r)
      pb[(r + 8 * hx) * 18 + lx] = p[r];
#pragma unroll
    for (int kc = 0; kc < 4; ++kc) {
      v2f pf = *(const v2f*)&pb[lx * 18 + kc * 4 + 2 * hx];
#pragma unroll
      for (int nt = 0; nt < 4; ++nt) {
        v2f b;
        b.x = V[(size_t)(kbase + kc * 4 + 2 * hx) * D_MODEL + hoff + nt * 16 + lx];
        b.y = V[(size_t)(kbase + kc * 4 + 2 * hx + 1) * D_MODEL + hoff + nt * 16 + lx];
        oacc[nt] = wmma4(pf, b, oacc[nt]);
      }
    }
  }
  // normalize and write out (row always has >=1 unmasked key -> lr > 0)
#pragma unroll
  for (int nt = 0; nt < 4; ++nt)
#pragma unroll
    for (int r = 0; r < 8; ++r)
      O[(size_t)(qbase + r + 8 * hx) * D_MODEL + hoff + nt * 16 + lx] =
          oacc[nt][r] / lr[r];
}

// ---------------------------------------------------------------------------
// SiLU(gate) * up -> gate (in place)
// ---------------------------------------------------------------------------
__global__ __launch_bounds__(256) void silu_mul_kernel(float* __restrict__ g,
                                                       const float* __restrict__ u) {
  const size_t i = (size_t)blockIdx.x * blockDim.x + threadIdx.x;
  const float gv = g[i];
  g[i] = (gv / (1.0f + __expf(-gv))) * u[i];
}

// ---------------------------------------------------------------------------
extern "C" void kernel_launch(void* const* d_in, const int* in_sizes, int n_in,
                              void* d_out, int out_size, void* d_ws, size_t ws_size,
                              hipStream_t stream) {
  const float* x       = (const float*)d_in[0];
  const float* w_q     = (const float*)d_in[1];
  const float* w_k     = (const float*)d_in[2];
  const float* w_v     = (const float*)d_in[3];
  const float* w_o     = (const float*)d_in[4];
  const float* w_gate  = (const float*)d_in[5];
  const float* w_up    = (const float*)d_in[6];
  const float* w_down  = (const float*)d_in[7];
  const float* w_norm1 = (const float*)d_in[8];
  const float* w_norm2 = (const float*)d_in[9];
  float* out = (float*)d_out;

  const size_t SD = (size_t)S_LEN * D_MODEL;   // 3,145,728
  const size_t SF = (size_t)S_LEN * F_FFN;     // 8,388,608
  float* f    = (float*)d_ws;
  float* xn   = f;            // rmsnorm output (reused as y after h)
  float* q    = xn + SD;
  float* k    = q + SD;
  float* v    = k + SD;
  float* attn = v + SD;
  float* h    = attn + SD;
  float* up   = h + SD;
  float* gate = q;            // reuse q/k/v region (3*SD >= SF) after attention

  const dim3 blk(256);
  const dim3 gD(D_MODEL / 64, S_LEN / 128);   // N=768 GEMMs
  const dim3 gF(F_FFN / 64, S_LEN / 128);     // N=2048 GEMMs

  // 1) y1 = rmsnorm(x) * w1
  rmsnorm_kernel<<<S_LEN, blk, 0, stream>>>(x, w_norm1, xn);
  // 2) Q,K,V projections
  gemm_wmma_f32<<<gD, blk, 0, stream>>>(xn, w_q, nullptr, q, S_LEN, D_MODEL, D_MODEL);
  gemm_wmma_f32<<<gD, blk, 0, stream>>>(xn, w_k, nullptr, k, S_LEN, D_MODEL, D_MODEL);
  gemm_wmma_f32<<<gD, blk, 0, stream>>>(xn, w_v, nullptr, v, S_LEN, D_MODEL, D_MODEL);
  // 3) RoPE in place
  rope_kernel<<<(S_LEN * NHEAD * 32) / 256, blk, 0, stream>>>(q, k);
  // 4) causal flash attention
  flash_attn_f32<<<dim3(S_LEN / 128, NHEAD), blk, 0, stream>>>(q, k, v, attn);
  // 5) h = x + attn @ w_o
  gemm_wmma_f32<<<gD, blk, 0, stream>>>(attn, w_o, x, h, S_LEN, D_MODEL, D_MODEL);
  // 6) y2 = rmsnorm(h) * w2
  rmsnorm_kernel<<<S_LEN, blk, 0, stream>>>(h, w_norm2, xn);
  // 7) gate/up projections
  gemm_wmma_f32<<<gF, blk, 0, stream>>>(xn, w_gate, nullptr, gate, S_LEN, F_FFN, D_MODEL);
  gemm_wmma_f32<<<gF, blk, 0, stream>>>(xn, w_up, nullptr, up, S_LEN, F_FFN, D_MODEL);
  // 8) act = silu(gate) * up
  silu_mul_kernel<<<SF / 256, blk, 0, stream>>>(gate, up);
  // 9) out = h + act @ w_down
  gemm_wmma_f32<<<gD, blk, 0, stream>>>(gate, w_down, h, out, S_LEN, D_MODEL, F_FFN);
}